// MSGNN_node_classification_54778012893401
// MI455X (gfx1250) — compile-verified
//
#include <hip/hip_runtime.h>
#include <math.h>

#define NN   100000
#define EE   1600000
#define TWO_PI_Q 1.57079632679489662f   // 2*pi*0.25

typedef __attribute__((ext_vector_type(2))) float v2f;
typedef __attribute__((ext_vector_type(8))) float v8f;

#define THREADS 256

// ---------------------------------------------------------------- utilities
__global__ void zero_f4_kernel(float4* __restrict__ p, int n4) {
    int t = blockIdx.x * blockDim.x + threadIdx.x;
    if (t < n4) p[t] = make_float4(0.f, 0.f, 0.f, 0.f);
}

__global__ void negcopy_f4_kernel(float4* __restrict__ d, const float4* __restrict__ s, int n4) {
    int t = blockIdx.x * blockDim.x + threadIdx.x;
    if (t < n4) {
        float4 v = s[t];
        d[t] = make_float4(-v.x, -v.y, -v.z, -v.w);
    }
}

__global__ void fill_bias_kernel(float* __restrict__ oR, float* __restrict__ oI,
                                 const float* __restrict__ b, int total) {
    int t = blockIdx.x * blockDim.x + threadIdx.x;
    if (t >= total) return;
    float bv = b[t & 31];
    oR[t] = bv;
    oI[t] = bv;
}

// ---------------------------------------------------------------- operator build
__global__ void degree_kernel(const int* __restrict__ src, const int* __restrict__ dst,
                              const float* __restrict__ w, float* __restrict__ deg) {
    int j = blockIdx.x * blockDim.x + threadIdx.x;
    if (j >= EE) return;
    float a = fabsf(0.5f * w[j]);
    atomicAdd(&deg[src[j]], a);
    atomicAdd(&deg[dst[j]], a);
}

__global__ void dinv_kernel(float* __restrict__ deg) {
    int i = blockIdx.x * blockDim.x + threadIdx.x;
    if (i >= NN) return;
    float d = deg[i];
    deg[i] = (d > 0.f) ? rsqrtf(fmaxf(d, 1e-12f)) : 0.f;
}

__global__ void hrhi_kernel(const int* __restrict__ src, const int* __restrict__ dst,
                            const float* __restrict__ w, const float* __restrict__ dinv,
                            float* __restrict__ hr, float* __restrict__ hi) {
    int j = blockIdx.x * blockDim.x + threadIdx.x;
    if (j >= EE) return;
    float wj = w[j];
    float a  = 0.5f * wj;
    float nrm = -(dinv[src[j]] * a * dinv[dst[j]]);
    float s, c;
    sincosf(TWO_PI_Q * wj, &s, &c);
    hr[j]      = nrm * c;   hi[j]      = nrm * s;   // entry (src->dst), theta = +
    hr[j + EE] = nrm * c;   hi[j + EE] = -nrm * s;  // entry (dst->src), theta = -
}

// ---------------------------------------------------------------- complex SpMM (segment_sum via L2 atomics)
// accumulates: p += scale * (h * u), complex; entry e<EE: row=src,col=dst ; else swapped
__global__ void spmm_kernel(const int* __restrict__ src, const int* __restrict__ dst,
                            const float* __restrict__ hr, const float* __restrict__ hi,
                            const float* __restrict__ ur, const float* __restrict__ ui,
                            float* __restrict__ pr, float* __restrict__ pi,
                            int F, int chunkShift, float scale) {
    int t = blockIdx.x * blockDim.x + threadIdx.x;
    int e = t >> chunkShift;
    if (e >= 2 * EE) return;
    int c = t & ((1 << chunkShift) - 1);
    int row, col;
    if (e < EE) { row = src[e]; col = dst[e]; }
    else        { row = dst[e - EE]; col = src[e - EE]; }
    float hre = hr[e] * scale;
    float him = hi[e] * scale;
    const float4 U = ((const float4*)(ur + col * F))[c];
    const float4 V = ((const float4*)(ui + col * F))[c];
    int base = row * F + (c << 2);
    atomicAdd(&pr[base + 0], hre * U.x - him * V.x);
    atomicAdd(&pr[base + 1], hre * U.y - him * V.y);
    atomicAdd(&pr[base + 2], hre * U.z - him * V.z);
    atomicAdd(&pr[base + 3], hre * U.w - him * V.w);
    atomicAdd(&pi[base + 0], hre * V.x + him * U.x);
    atomicAdd(&pi[base + 1], hre * V.y + him * U.y);
    atomicAdd(&pi[base + 2], hre * V.z + him * U.z);
    atomicAdd(&pi[base + 3], hre * V.w + him * U.w);
}

// ---------------------------------------------------------------- WMMA GEMM: out[N,32] += T[N,F] @ W[F,32]
// One wave per 16x16 C tile; real and imag share the B fragments.
// Fragment layouts per CDNA5 ISA 7.12.2:
//   A (16x4 f32): lane m=L&15, VGPR g -> K = k0 + 2*(L>>4) + g
//   B (4x16 f32): lane n=L&15, VGPR g -> K = k0 + 2*(L>>4) + g
//   C/D (16x16 f32): lane n=L&15, VGPR g -> M = g + 8*(L>>4)
__global__ void gemm_acc_kernel(const float* __restrict__ Tr, const float* __restrict__ Ti,
                                const float* __restrict__ W,
                                float* __restrict__ outR, float* __restrict__ outI,
                                int F, int numTilesM) {
    int gid  = blockIdx.x * blockDim.x + threadIdx.x;
    int wave = gid >> 5;
    if (wave >= numTilesM * 2) return;       // wave-uniform: EXEC stays all-1s for WMMA
    int lane = threadIdx.x & 31;
    int tm = wave >> 1;
    int tn = wave & 1;
    int mBase = tm * 16;
    int lo = lane & 15;
    int hi = lane >> 4;
    int n  = tn * 16 + lo;

    v8f cr, ci;
#pragma unroll
    for (int g = 0; g < 8; ++g) {
        int row = mBase + g + 8 * hi;
        cr[g] = outR[row * 32 + n];
        ci[g] = outI[row * 32 + n];
    }

    const float* trRow = Tr + (size_t)(mBase + lo) * F;
    const float* tiRow = Ti + (size_t)(mBase + lo) * F;
    for (int k0 = 0; k0 < F; k0 += 4) {
        int ka = k0 + 2 * hi;
        v2f ar, ai, b;
        ar.x = trRow[ka];     ar.y = trRow[ka + 1];
        ai.x = tiRow[ka];     ai.y = tiRow[ka + 1];
        b.x  = W[(ka + 0) * 32 + n];
        b.y  = W[(ka + 1) * 32 + n];
        cr = __builtin_amdgcn_wmma_f32_16x16x4_f32(false, ar, false, b, (short)0, cr, false, false);
        ci = __builtin_amdgcn_wmma_f32_16x16x4_f32(false, ai, false, b, (short)0, ci, false, false);
    }

#pragma unroll
    for (int g = 0; g < 8; ++g) {
        int row = mBase + g + 8 * hi;
        outR[row * 32 + n] = cr[g];
        outI[row * 32 + n] = ci[g];
    }
}

// ---------------------------------------------------------------- complex ReLU (mask by sign of real part)
__global__ void relu_kernel(const float* __restrict__ oR, const float* __restrict__ oI,
                            float* __restrict__ xr, float* __restrict__ xi, int total) {
    int t = blockIdx.x * blockDim.x + threadIdx.x;
    if (t >= total) return;
    float r = oR[t];
    float m = (r >= 0.f) ? 1.f : 0.f;
    xr[t] = r * m;
    xi[t] = oI[t] * m;
}

// ---------------------------------------------------------------- head: linear + log_softmax + argmax + softmax + L2-norm
__global__ void head_kernel(const float* __restrict__ xr, const float* __restrict__ xi,
                            const float* __restrict__ convW, const float* __restrict__ convB,
                            float* __restrict__ out) {
    int n = blockIdx.x * blockDim.x + threadIdx.x;
    if (n >= NN) return;
    float x[64];
#pragma unroll
    for (int f = 0; f < 32; ++f) x[f] = xr[n * 32 + f];
#pragma unroll
    for (int f = 0; f < 32; ++f) x[32 + f] = xi[n * 32 + f];

    float ss = 0.f;
#pragma unroll
    for (int f = 0; f < 64; ++f) ss += x[f] * x[f];
    float inv = 1.f / fmaxf(sqrtf(ss), 1e-12f);

    float logits[8];
#pragma unroll
    for (int c = 0; c < 8; ++c) logits[c] = convB[c];
    for (int f = 0; f < 64; ++f) {
        float xv = x[f];
#pragma unroll
        for (int c = 0; c < 8; ++c) logits[c] += xv * convW[f * 8 + c];
    }

    float mx = logits[0];
#pragma unroll
    for (int c = 1; c < 8; ++c) mx = fmaxf(mx, logits[c]);
    float se = 0.f;
#pragma unroll
    for (int c = 0; c < 8; ++c) se += __expf(logits[c] - mx);
    float lse = mx + __logf(se);

    int am = 0; float best = logits[0];
#pragma unroll
    for (int c = 1; c < 8; ++c) if (logits[c] > best) { best = logits[c]; am = c; }

    float* zout = out;                              // [N,64]
    float* lout = out + (size_t)NN * 64;            // [N,8]  log_softmax
    float* pout = lout + (size_t)NN * 8;            // [N]    argmax (as float)
    float* prob = pout + NN;                        // [N,8]  softmax
#pragma unroll
    for (int f = 0; f < 64; ++f) zout[(size_t)n * 64 + f] = x[f] * inv;
#pragma unroll
    for (int c = 0; c < 8; ++c) {
        float lsm = logits[c] - lse;
        lout[n * 8 + c] = lsm;
        prob[n * 8 + c] = __expf(lsm);
    }
    pout[n] = (float)am;
}

// ---------------------------------------------------------------- orchestration
static inline int cdiv(int a, int b) { return (a + b - 1) / b; }

extern "C" void kernel_launch(void* const* d_in, const int* in_sizes, int n_in,
                              void* d_out, int out_size, void* d_ws, size_t ws_size,
                              hipStream_t stream) {
    const float* real  = (const float*)d_in[0];
    const float* imag  = (const float*)d_in[1];
    const int*   ei    = (const int*)d_in[2];
    const float* ew    = (const float*)d_in[3];
    const float* W0    = (const float*)d_in[4];
    const float* b0    = (const float*)d_in[5];
    const float* W1    = (const float*)d_in[6];
    const float* b1    = (const float*)d_in[7];
    const float* W2    = (const float*)d_in[8];
    const float* b2    = (const float*)d_in[9];
    const float* convW = (const float*)d_in[10];
    const float* convB = (const float*)d_in[11];

    const int* src = ei;
    const int* dst = ei + EE;

    float* ws  = (float*)d_ws;
    float* hr  = ws;                  // 2E
    float* hi  = hr + 2 * EE;         // 2E
    float* deg = hi + 2 * EE;         // N (deg -> dinv in place)
    float* Ar  = deg + NN;            // N*64
    float* Ai  = Ar + NN * 64;        // N*64
    float* Br  = Ai + NN * 64;        // N*64
    float* Bi  = Br + NN * 64;        // N*64
    float* xr  = Bi + NN * 64;        // N*32
    float* xi  = xr + NN * 32;        // N*32
    float* oR  = xi + NN * 32;        // N*32
    float* oI  = oR + NN * 32;        // N*32

    const int numTilesM = NN / 16;    // 6250
    const int gemmBlocks = cdiv(numTilesM * 2 * 32, THREADS);

    // ---- operator build
    zero_f4_kernel<<<cdiv(NN / 4, THREADS), THREADS, 0, stream>>>((float4*)deg, NN / 4);
    degree_kernel<<<cdiv(EE, THREADS), THREADS, 0, stream>>>(src, dst, ew, deg);
    dinv_kernel<<<cdiv(NN, THREADS), THREADS, 0, stream>>>(deg);
    hrhi_kernel<<<cdiv(EE, THREADS), THREADS, 0, stream>>>(src, dst, ew, deg, hr, hi);

    const float* pxr = real;
    const float* pxi = imag;
    const float* Ws[3] = {W0, W1, W2};
    const float* bs[3] = {b0, b1, b2};

    for (int layer = 0; layer < 3; ++layer) {
        const int F = (layer == 0) ? 64 : 32;
        const int cs = (F == 64) ? 4 : 3;                 // chunkShift (float4 chunks)
        const size_t wstride = (size_t)F * 32;            // per-k weight matrix size
        const int nf4 = NN * F / 4;
        const int spmmBlocks = cdiv(2 * EE * (F / 4), THREADS);

        // out = bias
        fill_bias_kernel<<<cdiv(NN * 32, THREADS), THREADS, 0, stream>>>(oR, oI, bs[layer], NN * 32);
        // out += t0 @ W[0]
        gemm_acc_kernel<<<gemmBlocks, THREADS, 0, stream>>>(pxr, pxi, Ws[layer] + 0 * wstride, oR, oI, F, numTilesM);
        // t1 = prop(t0)
        zero_f4_kernel<<<cdiv(nf4, THREADS), THREADS, 0, stream>>>((float4*)Ar, nf4);
        zero_f4_kernel<<<cdiv(nf4, THREADS), THREADS, 0, stream>>>((float4*)Ai, nf4);
        spmm_kernel<<<spmmBlocks, THREADS, 0, stream>>>(src, dst, hr, hi, pxr, pxi, Ar, Ai, F, cs, 1.0f);
        // out += t1 @ W[1]
        gemm_acc_kernel<<<gemmBlocks, THREADS, 0, stream>>>(Ar, Ai, Ws[layer] + 1 * wstride, oR, oI, F, numTilesM);
        // t2 = 2*prop(t1) - t0  (init B = -t0, accumulate with scale 2)
        negcopy_f4_kernel<<<cdiv(nf4, THREADS), THREADS, 0, stream>>>((float4*)Br, (const float4*)pxr, nf4);
        negcopy_f4_kernel<<<cdiv(nf4, THREADS), THREADS, 0, stream>>>((float4*)Bi, (const float4*)pxi, nf4);
        spmm_kernel<<<spmmBlocks, THREADS, 0, stream>>>(src, dst, hr, hi, Ar, Ai, Br, Bi, F, cs, 2.0f);
        // out += t2 @ W[2]
        gemm_acc_kernel<<<gemmBlocks, THREADS, 0, stream>>>(Br, Bi, Ws[layer] + 2 * wstride, oR, oI, F, numTilesM);
        // complex ReLU -> next-layer input
        relu_kernel<<<cdiv(NN * 32, THREADS), THREADS, 0, stream>>>(oR, oI, xr, xi, NN * 32);
        pxr = xr;
        pxi = xi;
    }

    head_kernel<<<cdiv(NN, THREADS), THREADS, 0, stream>>>(xr, xi, convW, convB, (float*)d_out);
}